// GCN2_14370960572518
// MI455X (gfx1250) — compile-verified
//
#include <hip/hip_runtime.h>
#include <hip/hip_bf16.h>
#include <math.h>

// ---------------- problem constants (match reference) ----------------
#define IN_CH      256
#define HID        128
#define OUT_CH     40
#define NUM_LAYERS 8

typedef __attribute__((ext_vector_type(2))) float v2f;
typedef __attribute__((ext_vector_type(8))) float v8f;

// padded LDS row strides (avoid 16-way bank conflicts: 256,128 % 64banks == 0)
// both remain multiples of 4 so float4 (B128) stores stay 16B-aligned
#define ALDA1 (IN_CH + 4)   // lin1 A strip row stride
#define ALDA2 (HID + 4)     // conv A strip row stride

// ---------------------------------------------------------------------
// degree / normalization
// ---------------------------------------------------------------------
__global__ void gcn2_deg_init(float* __restrict__ deg, int n) {
    int i = blockIdx.x * blockDim.x + threadIdx.x;
    if (i < n) deg[i] = 1.0f;                 // self-loop contribution
}

__global__ void gcn2_deg_accum(const int* __restrict__ cols, float* __restrict__ deg, int e) {
    int i = blockIdx.x * blockDim.x + threadIdx.x;
    if (i < e) atomicAdd(&deg[cols[i]], 1.0f);
}

__global__ void gcn2_dinv(float* __restrict__ deg, int n) {
    int i = blockIdx.x * blockDim.x + threadIdx.x;
    if (i < n) deg[i] = rsqrtf(deg[i]);       // deg >= 1 always (self loops)
}

// ---------------------------------------------------------------------
// lin1: h = relu(x @ W1 + b1); x0 = h.   [n x 256] @ [256 x 128]
// 8 waves/block, each wave owns one 16-col tile; WMMA f32 16x16x4.
// ---------------------------------------------------------------------
__global__ __launch_bounds__(256) void gcn2_lin1(
    const float* __restrict__ x, const float* __restrict__ w,
    const float* __restrict__ bias, float* __restrict__ h,
    float* __restrict__ x0, int n)
{
    __shared__ float wlds[IN_CH * HID];       // 128 KB weights, block-resident
    __shared__ float alds[16 * ALDA1];        // 16-row input strip (padded)

    const int tid = threadIdx.x;
    for (int i = tid; i < IN_CH * HID; i += 256) wlds[i] = w[i];

    const int wave = tid >> 5;
    const int lane = tid & 31;
    const int n0   = wave * 16;               // output column tile
    const int lr   = lane & 15;
    const int lh   = lane >> 4;

    const float4* x4 = reinterpret_cast<const float4*>(x);

    const int nstrips = (n + 15) >> 4;
    for (int strip = blockIdx.x; strip < nstrips; strip += gridDim.x) {
        const int m0 = strip << 4;
        __syncthreads();                      // alds safe to overwrite
        // stage 16 x 256 strip as float4 (B128 loads)
        for (int i = tid; i < 16 * (IN_CH / 4); i += 256) {
            int r = i >> 6, c4 = i & 63;
            int gr = m0 + r;
            float4 v = {0.0f, 0.0f, 0.0f, 0.0f};
            if (gr < n) v = x4[(size_t)gr * (IN_CH / 4) + c4];
            *reinterpret_cast<float4*>(&alds[r * ALDA1 + c4 * 4]) = v;
        }
        __syncthreads();

        // prefetch the rows of the NEXT strip this block will process
        {
            int pr = ((strip + gridDim.x) << 4) + (tid & 15);
            if (tid < 16 && pr < n)
                __builtin_prefetch(&x[(size_t)pr * IN_CH], 0, 1);
        }

        v8f acc = {};
        for (int k0 = 0; k0 < IN_CH; k0 += 4) {
            const int ka = k0 + 2 * lh;       // ISA 7.12.2 f32 A/B lane layout
            v2f a, b;
            a.x = alds[lr * ALDA1 + ka];
            a.y = alds[lr * ALDA1 + ka + 1];
            b.x = wlds[ka * HID + n0 + lr];
            b.y = wlds[(ka + 1) * HID + n0 + lr];
            acc = __builtin_amdgcn_wmma_f32_16x16x4_f32(
                false, a, false, b, (short)0, acc, false, false);
        }

        const float bv = bias[n0 + lr];
        if (m0 + 15 < n) {                    // full strip: no EXEC churn
            #pragma unroll
            for (int v = 0; v < 8; ++v) {     // D: VGPR v -> rows v, v+8
                int r = m0 + v + 8 * lh;
                float val = fmaxf(acc[v] + bv, 0.0f);
                size_t idx = (size_t)r * HID + n0 + lr;
                h[idx]  = val;
                x0[idx] = val;
            }
        } else {
            #pragma unroll
            for (int v = 0; v < 8; ++v) {
                int r = m0 + v + 8 * lh;
                if (r < n) {
                    float val = fmaxf(acc[v] + bv, 0.0f);
                    size_t idx = (size_t)r * HID + n0 + lr;
                    h[idx]  = val;
                    x0[idx] = val;
                }
            }
        }
    }
}

// ---------------------------------------------------------------------
// agg init: agg[i] = dinv[i]^2 * h[i]   (self-loop term, also zeros rest)
// ---------------------------------------------------------------------
__global__ void gcn2_agg_init(const float* __restrict__ dinv, const float* __restrict__ h,
                              float* __restrict__ agg, int n)
{
    int idx = blockIdx.x * blockDim.x + threadIdx.x;     // over n * HID/4
    if (idx < n * (HID / 4)) {
        int r = idx >> 5;
        float d = dinv[r];
        d *= d;
        float4 v = reinterpret_cast<const float4*>(h)[idx];
        float4 o = {d * v.x, d * v.y, d * v.z, d * v.w};
        reinterpret_cast<float4*>(agg)[idx] = o;
    }
}

// ---------------------------------------------------------------------
// edge scatter: agg[col] += dinv[row]*dinv[col] * h[row]
// one wave per edge: 32 lanes x float4 = 128 channels; 512B coalesced row
// read, 128 f32 L2 atomics per edge. 8 edges per 256-thread block.
// ---------------------------------------------------------------------
__global__ __launch_bounds__(256) void gcn2_scatter(
    const int* __restrict__ rows, const int* __restrict__ cols,
    const float* __restrict__ dinv, const float* __restrict__ h,
    float* __restrict__ agg, int e)
{
    int ed   = blockIdx.x * 8 + (threadIdx.x >> 5);
    int lane = threadIdx.x & 31;
    if (ed < e) {
        int r = rows[ed], cl = cols[ed];
        float w = dinv[r] * dinv[cl];
        float4 hv = *reinterpret_cast<const float4*>(&h[(size_t)r * HID + lane * 4]);
        float* dst = &agg[(size_t)cl * HID + lane * 4];
        atomicAdd(dst + 0, w * hv.x);
        atomicAdd(dst + 1, w * hv.y);
        atomicAdd(dst + 2, w * hv.z);
        atomicAdd(dst + 3, w * hv.w);
    }
}

// ---------------------------------------------------------------------
// conv layer: hc = 0.5*agg + 0.5*x0 ;  h = relu((1-beta)*hc + beta*(hc@W))
// fused: hc built while staging the A strip; identity term read back from LDS
// ---------------------------------------------------------------------
__global__ __launch_bounds__(256) void gcn2_conv(
    const float* __restrict__ agg, const float* __restrict__ x0,
    const float* __restrict__ w, float* __restrict__ h, float beta, int n)
{
    __shared__ float wlds[HID * HID];         // 64 KB weights
    __shared__ float alds[16 * ALDA2];        // hc strip (padded)

    const int tid = threadIdx.x;
    for (int i = tid; i < HID * HID; i += 256) wlds[i] = w[i];

    const int wave = tid >> 5;
    const int lane = tid & 31;
    const int n0   = wave * 16;
    const int lr   = lane & 15;
    const int lh   = lane >> 4;
    const float ob = 1.0f - beta;

    const float4* agg4 = reinterpret_cast<const float4*>(agg);
    const float4* x04  = reinterpret_cast<const float4*>(x0);

    const int nstrips = (n + 15) >> 4;
    for (int strip = blockIdx.x; strip < nstrips; strip += gridDim.x) {
        const int m0 = strip << 4;
        __syncthreads();
        // stage hc = 0.5*agg + 0.5*x0 strip as float4
        for (int i = tid; i < 16 * (HID / 4); i += 256) {
            int r = i >> 5, c4 = i & 31;
            int gr = m0 + r;
            float4 v = {0.0f, 0.0f, 0.0f, 0.0f};
            if (gr < n) {
                float4 a4 = agg4[(size_t)gr * (HID / 4) + c4];
                float4 b4 = x04[(size_t)gr * (HID / 4) + c4];
                v = {0.5f * (a4.x + b4.x), 0.5f * (a4.y + b4.y),
                     0.5f * (a4.z + b4.z), 0.5f * (a4.w + b4.w)};
            }
            *reinterpret_cast<float4*>(&alds[r * ALDA2 + c4 * 4]) = v;
        }
        __syncthreads();

        // prefetch next strip's inputs
        {
            int pr = ((strip + gridDim.x) << 4) + (tid & 15);
            if (tid < 32 && pr < n) {
                if (tid < 16) __builtin_prefetch(&agg[(size_t)pr * HID], 0, 1);
                else          __builtin_prefetch(&x0[(size_t)pr * HID], 0, 1);
            }
        }

        v8f acc = {};
        for (int k0 = 0; k0 < HID; k0 += 4) {
            const int ka = k0 + 2 * lh;
            v2f a, b;
            a.x = alds[lr * ALDA2 + ka];
            a.y = alds[lr * ALDA2 + ka + 1];
            b.x = wlds[ka * HID + n0 + lr];
            b.y = wlds[(ka + 1) * HID + n0 + lr];
            acc = __builtin_amdgcn_wmma_f32_16x16x4_f32(
                false, a, false, b, (short)0, acc, false, false);
        }

        if (m0 + 15 < n) {                    // full strip fast path
            #pragma unroll
            for (int v = 0; v < 8; ++v) {
                int rl = v + 8 * lh;
                float hc  = alds[rl * ALDA2 + n0 + lr];
                float val = fmaxf(ob * hc + beta * acc[v], 0.0f);
                h[(size_t)(m0 + rl) * HID + n0 + lr] = val;
            }
        } else {
            #pragma unroll
            for (int v = 0; v < 8; ++v) {
                int rl = v + 8 * lh;
                int r  = m0 + rl;
                if (r < n) {
                    float hc  = alds[rl * ALDA2 + n0 + lr];
                    float val = fmaxf(ob * hc + beta * acc[v], 0.0f);
                    h[(size_t)r * HID + n0 + lr] = val;
                }
            }
        }
    }
}

// ---------------------------------------------------------------------
// lin2 + log_softmax, one 64-thread block per node
// ---------------------------------------------------------------------
__global__ __launch_bounds__(64) void gcn2_out(
    const float* __restrict__ h, const float* __restrict__ w2,
    const float* __restrict__ b2, float* __restrict__ out, int n)
{
    __shared__ float hl[HID];
    __shared__ float o[OUT_CH];
    __shared__ float red[2];

    int node = blockIdx.x;
    if (node >= n) return;                    // uniform per block
    int t = threadIdx.x;
    hl[t]      = h[(size_t)node * HID + t];
    hl[t + 64] = h[(size_t)node * HID + 64 + t];
    __syncthreads();

    if (t < OUT_CH) {
        float s = b2[t];
        #pragma unroll 4
        for (int k = 0; k < HID; ++k) s += hl[k] * w2[k * OUT_CH + t];
        o[t] = s;
    }
    __syncthreads();
    if (t == 0) {
        float m = -INFINITY;
        for (int i = 0; i < OUT_CH; ++i) m = fmaxf(m, o[i]);
        float s = 0.0f;
        for (int i = 0; i < OUT_CH; ++i) s += expf(o[i] - m);
        red[0] = m;
        red[1] = logf(s);
    }
    __syncthreads();
    if (t < OUT_CH) out[(size_t)node * OUT_CH + t] = o[t] - red[0] - red[1];
}

// ---------------------------------------------------------------------
extern "C" void kernel_launch(void* const* d_in, const int* in_sizes, int n_in,
                              void* d_out, int out_size, void* d_ws, size_t ws_size,
                              hipStream_t stream)
{
    const float* x    = (const float*)d_in[0];
    const int*   ei   = (const int*)  d_in[1];
    const float* w1   = (const float*)d_in[2];
    const float* b1   = (const float*)d_in[3];
    const float* cw   = (const float*)d_in[4];
    const float* w2   = (const float*)d_in[5];
    const float* b2   = (const float*)d_in[6];

    const int n = in_sizes[0] / IN_CH;        // 100000
    const int e = in_sizes[1] / 2;            // 1600000
    const int* rows = ei;                     // edge_index[0] = source
    const int* cols = ei + e;                 // edge_index[1] = target

    // workspace layout (floats): dinv[n] | h[n*HID] | x0[n*HID] | agg[n*HID]
    float* ws   = (float*)d_ws;
    float* dinv = ws;
    float* h    = ws + n;
    float* x0   = h  + (size_t)n * HID;
    float* agg  = x0 + (size_t)n * HID;

    // 1) symmetric GCN normalization
    gcn2_deg_init <<<(n + 255) / 256, 256, 0, stream>>>(dinv, n);
    gcn2_deg_accum<<<(e + 255) / 256, 256, 0, stream>>>(cols, dinv, e);
    gcn2_dinv     <<<(n + 255) / 256, 256, 0, stream>>>(dinv, n);

    // 2) input projection (WMMA GEMM) + relu, duplicated into x0
    gcn2_lin1<<<512, 256, 0, stream>>>(x, w1, b1, h, x0, n);

    // 3) 8 GCNII layers
    for (int l = 0; l < NUM_LAYERS; ++l) {
        float beta = logf(1.0f / (float)(l + 1) + 1.0f);   // THETA = 1
        gcn2_agg_init<<<(n * (HID / 4) + 255) / 256, 256, 0, stream>>>(dinv, h, agg, n);
        gcn2_scatter <<<(e + 7) / 8, 256, 0, stream>>>(rows, cols, dinv, h, agg, e);
        gcn2_conv    <<<512, 256, 0, stream>>>(agg, x0, cw + (size_t)l * HID * HID, h, beta, n);
    }

    // 4) output head + log_softmax
    gcn2_out<<<n, 64, 0, stream>>>(h, w2, b2, (float*)d_out, n);
}